// GNN_28587302322220
// MI455X (gfx1250) — compile-verified
//
#include <hip/hip_runtime.h>

typedef __attribute__((ext_vector_type(2))) float v2f;
typedef __attribute__((ext_vector_type(8))) float v8f;

// ---------------------------------------------------------------------------
// Degree: deg[i] = #incoming edges (segment over col); +1 self loop applied in
// the norm kernel.
// ---------------------------------------------------------------------------
__global__ void gcn_deg_kernel(const int* __restrict__ col,
                               unsigned int* __restrict__ deg, int E) {
  int e = blockIdx.x * blockDim.x + threadIdx.x;
  if (e < E) atomicAdd(&deg[col[e]], 1u);
}

__global__ void gcn_norm_kernel(const unsigned int* __restrict__ deg,
                                float* __restrict__ dinv,
                                float* __restrict__ deginv, int N) {
  int i = blockIdx.x * blockDim.x + threadIdx.x;
  if (i < N) {
    float d = (float)deg[i] + 1.0f;
    dinv[i]   = rsqrtf(d);
    deginv[i] = 1.0f / d;
  }
}

// ---------------------------------------------------------------------------
// hW = h_in @ W via V_WMMA_F32_16X16X4_F32.  One wave handles 16 nodes.
//  A (16x4, f32): lanes 0-15 hold {K=0,K=1} = the two input features,
//                 lanes 16-31 hold {K=2,K=3} = zero padding.
//  B (4x16, f32): lanes 0-15 hold rows K=0,K=1 of W (cols N=lane, zero for
//                 N>=dout), lanes 16-31 zero.
//  D (16x16,f32): element (M,N) lives in lane N (M<8) / lane N+16 (M>=8),
//                 VGPR M%8 -> only lanes 0..dout-1 and 16..16+dout-1 store.
// Also zeroes the agg buffer for this tile (no per-layer memset needed).
// Full-tile fast path keeps the stores branch-free for N % 16 == 0.
// ---------------------------------------------------------------------------
__global__ void gcn_transform_kernel(const float* __restrict__ hin,
                                     const float* __restrict__ W, int dout,
                                     float* __restrict__ hW,
                                     float* __restrict__ agg, int N) {
  const int lane = threadIdx.x & 31;
  const int wave = threadIdx.x >> 5;
  const int tile = blockIdx.x * (blockDim.x >> 5) + wave;
  const int base = tile * 16;
  if (base >= N) return;  // wave-uniform: EXEC stays all-1s for the WMMA
  const bool full = (base + 16 <= N);  // wave-uniform

  v2f a = {0.0f, 0.0f};
  if (lane < 16) {
    int node = base + lane;
    if (full || node < N) a = *(const v2f*)(hin + (size_t)node * 2);
  }
  v2f b = {0.0f, 0.0f};
  if (lane < dout) {  // implies lane < 16; W is row-major [din=2][dout]
    b.x = W[lane];
    b.y = W[dout + lane];
  }
  v8f c = {0.0f, 0.0f, 0.0f, 0.0f, 0.0f, 0.0f, 0.0f, 0.0f};
  v8f d = __builtin_amdgcn_wmma_f32_16x16x4_f32(
      /*neg_a=*/false, a, /*neg_b=*/false, b,
      /*c_mod=*/(short)0, c, /*reuse_a=*/false, /*reuse_b=*/false);

  // Zero the aggregation buffer for this tile (16 nodes * 2 comps = 32 lanes).
  {
    size_t idx = (size_t)base * 2 + lane;
    if (full || idx < (size_t)N * 2) agg[idx] = 0.0f;
  }

  const int n = lane & 15;
  const int mbase = (lane >> 4) * 8;
  if (n < dout) {
    float* o = hW + (size_t)(base + mbase) * 2 + n;
    if (full) {
#pragma unroll
      for (int m = 0; m < 8; ++m) o[2 * m] = d[m];
    } else {
#pragma unroll
      for (int m = 0; m < 8; ++m)
        if (base + mbase + m < N) o[2 * m] = d[m];
    }
  }
}

// ---------------------------------------------------------------------------
// Edge scatter: agg[col] += hW[row] * dinv[row] * dinv[col].
// Node arrays (hW, agg, dinv <= 8 MB) are L2-resident; index streams come
// from HBM.  hW gathered as a single b64 load for DOUT=2.
// ---------------------------------------------------------------------------
template <int DOUT>
__global__ void gcn_edge_kernel(const int* __restrict__ row,
                                const int* __restrict__ col,
                                const float* __restrict__ dinv,
                                const float* __restrict__ hW,
                                float* __restrict__ agg, int E) {
  int e = blockIdx.x * blockDim.x + threadIdx.x;
  if (e >= E) return;
  int r = row[e];
  int cI = col[e];
  float nrm = dinv[r] * dinv[cI];
  if (DOUT == 2) {
    v2f h = *(const v2f*)(hW + (size_t)r * 2);  // one b64 gather
    unsafeAtomicAdd(&agg[(size_t)cI * 2], h.x * nrm);
    unsafeAtomicAdd(&agg[(size_t)cI * 2 + 1], h.y * nrm);
  } else {
    float h0 = hW[(size_t)r * 2];
    unsafeAtomicAdd(&agg[(size_t)cI * 2], h0 * nrm);
  }
}

// ---------------------------------------------------------------------------
// out = agg + hW * deg_inv + bias ; optional ReLU.
// Intermediate layers write stride-2 node features (b64); final layer stride-1.
// ---------------------------------------------------------------------------
template <int DOUT, bool RELU>
__global__ void gcn_combine_kernel(const float* __restrict__ agg,
                                   const float* __restrict__ hW,
                                   const float* __restrict__ deginv,
                                   const float* __restrict__ bias,
                                   float* __restrict__ out, int N) {
  int i = blockIdx.x * blockDim.x + threadIdx.x;
  if (i >= N) return;
  float di = deginv[i];
  if (DOUT == 2) {
    v2f ag = *(const v2f*)(agg + (size_t)i * 2);
    v2f hw = *(const v2f*)(hW + (size_t)i * 2);
    v2f v;
    v.x = ag.x + hw.x * di + bias[0];
    v.y = ag.y + hw.y * di + bias[1];
    if (RELU) {
      v.x = fmaxf(v.x, 0.0f);
      v.y = fmaxf(v.y, 0.0f);
    }
    *(v2f*)(out + (size_t)i * 2) = v;
  } else {
    float v = agg[(size_t)i * 2] + hW[(size_t)i * 2] * di + bias[0];
    if (RELU) v = fmaxf(v, 0.0f);
    out[i] = v;
  }
}

extern "C" void kernel_launch(void* const* d_in, const int* in_sizes, int n_in,
                              void* d_out, int out_size, void* d_ws,
                              size_t ws_size, hipStream_t stream) {
  (void)n_in;
  (void)out_size;
  (void)ws_size;

  const float* x = (const float*)d_in[0];
  const int* ei  = (const int*)d_in[1];  // [2, E] int32 per harness convention
  const float* W[5];
  const float* Bv[5];
  for (int l = 0; l < 5; ++l) {
    W[l]  = (const float*)d_in[2 + 2 * l];
    Bv[l] = (const float*)d_in[3 + 2 * l];
  }
  const int N = in_sizes[0] / 2;
  const int E = in_sizes[1] / 2;
  const int* row = ei;
  const int* col = ei + E;

  // Workspace layout (~44 MB for N=1M).
  char* p = (char*)d_ws;
  float* dinv = (float*)p;   p += (size_t)N * sizeof(float);
  float* deginv = (float*)p; p += (size_t)N * sizeof(float);
  float* hA = (float*)p;     p += (size_t)N * 2 * sizeof(float);
  float* hB = (float*)p;     p += (size_t)N * 2 * sizeof(float);
  float* hW = (float*)p;     p += (size_t)N * 2 * sizeof(float);
  float* agg = (float*)p;    p += (size_t)N * 2 * sizeof(float);
  unsigned int* deg = (unsigned int*)p;

  const int TB = 256;  // 8 waves/block on wave32
  const int eblocks = (E + TB - 1) / TB;
  const int nblocks = (N + TB - 1) / TB;
  const int tiles = (N + 15) / 16;
  const int tblocks = (tiles + (TB / 32) - 1) / (TB / 32);

  hipMemsetAsync(deg, 0, (size_t)N * sizeof(unsigned int), stream);
  gcn_deg_kernel<<<eblocks, TB, 0, stream>>>(col, deg, E);
  gcn_norm_kernel<<<nblocks, TB, 0, stream>>>(deg, dinv, deginv, N);

  const float* hin = x;
  float* hbuf[2] = {hA, hB};
  for (int l = 0; l < 5; ++l) {
    const int dout = (l == 4) ? 1 : 2;
    gcn_transform_kernel<<<tblocks, TB, 0, stream>>>(hin, W[l], dout, hW, agg, N);
    if (dout == 2)
      gcn_edge_kernel<2><<<eblocks, TB, 0, stream>>>(row, col, dinv, hW, agg, E);
    else
      gcn_edge_kernel<1><<<eblocks, TB, 0, stream>>>(row, col, dinv, hW, agg, E);
    if (l < 4) {
      gcn_combine_kernel<2, true><<<nblocks, TB, 0, stream>>>(
          agg, hW, deginv, Bv[l], hbuf[l & 1], N);
      hin = hbuf[l & 1];
    } else {
      gcn_combine_kernel<1, false><<<nblocks, TB, 0, stream>>>(
          agg, hW, deginv, Bv[l], (float*)d_out, N);
    }
  }
}